// DictionaryLearningTokenized_23081154249057
// MI455X (gfx1250) — compile-verified
//
#include <hip/hip_runtime.h>
#include <math.h>

typedef __attribute__((ext_vector_type(2))) float v2f;
typedef __attribute__((ext_vector_type(8))) float v8f;

#define NUM_ATOMS   256
#define DIM         64
#define KSPARSE     4
#define SIG_PER_WG  64      // one 128-thread (4-wave) WG handles 64 signals
#define HW          4096    // 64*64 spatial positions per batch image
#define N_SIGNALS   (32*64*64)
#define OUT_ZQ_ELEMS (32*64*64*64)

// LDS row strides (floats): even (8B alignment for b64) and bank-spreading
#define XS_STRIDE   66      // Xs[signal][dim]
#define DLS_STRIDE  66      // Dls[atom][dim]
#define HBT_STRIDE  68      // HbT[atom][signal], 16B-aligned rows

// ---------------------------------------------------------------------------
// Kernel 1: normalize dictionary columns, build Gram matrix G = Dn^T Dn,
// emit transposed dictionary DnT[atom][dim], zero the loss accumulator.
// One 256-thread workgroup (one thread per atom).
// ---------------------------------------------------------------------------
__global__ __launch_bounds__(256) void dl_prep_kernel(
    const float* __restrict__ dict,   // (64, 256) row-major
    float* __restrict__ DnT,          // ws: (256, 64)  [atom][dim]
    float* __restrict__ G,            // ws: (256, 256)
    float* __restrict__ out_loss)
{
  extern __shared__ float DnL[];      // [atom][dim] = 256*64 floats
  const int j = threadIdx.x;          // atom index

  float n2 = 0.f;
  for (int c = 0; c < DIM; ++c) {
    float v = dict[c * NUM_ATOMS + j];
    n2 += v * v;
  }
  const float inv = 1.0f / fmaxf(sqrtf(n2), 1e-10f);
  for (int c = 0; c < DIM; ++c) {
    float v = dict[c * NUM_ATOMS + j] * inv;
    DnL[j * DIM + c] = v;
    DnT[j * DIM + c] = v;
  }
  __syncthreads();

  // Gram row j (256 dots of length 64 — trivial)
  for (int i = 0; i < NUM_ATOMS; ++i) {
    float s = 0.f;
    for (int c = 0; c < DIM; ++c)
      s += DnL[j * DIM + c] * DnL[i * DIM + c];
    G[j * NUM_ATOMS + i] = s;
  }
  if (j == 0) *out_loss = 0.f;
}

// ---------------------------------------------------------------------------
// Kernel 2: fused  h_bar = X^T Dn  (fp32 WMMA, slab kept in LDS)  +
// batch-OMP (K=4) + mu-law quantize + reconstruction + loss.
// 128 threads (4 waves) per WG; each wave WMMAs a 16x256 h_bar slab with
// A fragments held in VGPRs; one thread per signal runs the OMP solver.
// ---------------------------------------------------------------------------
__global__ __launch_bounds__(128) void dl_omp_kernel(
    const float* __restrict__ ze,     // (32, 64, 64, 64)
    const float* __restrict__ DnT,    // (256, 64) [atom][dim]
    const float* __restrict__ G,      // (256, 256)
    float* __restrict__ out,          // z_q (32,64,64,64) flat
    float* __restrict__ out_loss)
{
  extern __shared__ float smem[];
  float* Xs   = smem;                               // [64][XS_STRIDE]
  float* HbT  = Xs   + SIG_PER_WG * XS_STRIDE;      // [256][HBT_STRIDE]
  float* Dls  = HbT  + NUM_ATOMS * HBT_STRIDE;      // [256][DLS_STRIDE]
  float* red  = Dls  + NUM_ATOMS * DLS_STRIDE;      // [128] loss partials

  const int tid = threadIdx.x;
  const int s0  = blockIdx.x * SIG_PER_WG;          // first signal of tile
  const int b   = s0 >> 12;                         // batch image
  const int r0  = s0 & (HW - 1);                    // spatial offset
  const long zbase = (long)b * DIM * HW + r0;

  // --- stage signal tile transposed: Xs[t][c] (global reads coalesced) ---
  for (int f = tid; f < DIM * SIG_PER_WG; f += 128) {
    const int c = f >> 6, t = f & 63;
    Xs[t * XS_STRIDE + c] = ze[zbase + (long)c * HW + t];
  }
  // --- stage transposed dictionary: Dls[atom][c] (straight coalesced copy) ---
  for (int f = tid; f < DIM * NUM_ATOMS; f += 128) {
    const int a = f >> 6, c = f & 63;
    Dls[a * DLS_STRIDE + c] = DnT[f];
  }
  __syncthreads();

  // --- fp32 WMMA: wave w computes h_bar rows [w*16, w*16+16) x 256 atoms ---
  // A (16x4 f32): lanes 0-15 -> K={k0,k0+1}, lanes 16-31 -> K={k0+2,k0+3}
  const int lane = tid & 31;
  const int wv   = tid >> 5;
  const int l16  = lane & 15;
  const int hi   = (lane >= 16);
  const int koff = hi ? 2 : 0;

  // hoist all 16 A fragments (wave's 16x64 tile = 32 VGPRs) into registers
  v2f afrag[16];
  {
    const int arow = (wv * 16 + l16) * XS_STRIDE + koff;
#pragma unroll
    for (int kk = 0; kk < 16; ++kk)
      afrag[kk] = *(const v2f*)&Xs[arow + kk * 4];
  }

  for (int nt = 0; nt < 16; ++nt) {                 // 16 atom tiles of 16
    v8f acc = {0.f, 0.f, 0.f, 0.f, 0.f, 0.f, 0.f, 0.f};
    const int brow = (nt * 16 + l16) * DLS_STRIDE + koff;
#pragma unroll
    for (int kk = 0; kk < 16; ++kk) {               // K = 64 in steps of 4
      v2f bf = *(const v2f*)&Dls[brow + kk * 4];
      acc = __builtin_amdgcn_wmma_f32_16x16x4_f32(
          /*neg_a=*/false, afrag[kk], /*neg_b=*/false, bf,
          /*c_mod=*/(short)0, acc, /*reuse_a=*/false, /*reuse_b=*/false);
    }
    // C/D layout: VGPR r, lanes 0-15 -> M=r, lanes 16-31 -> M=r+8, N=l16.
    // Transposed store: lane's 8 M-values are 8 consecutive floats in HbT.
    float* dst = &HbT[(nt * 16 + l16) * HBT_STRIDE + wv * 16 + (hi ? 8 : 0)];
    float4 lo4 = {acc[0], acc[1], acc[2], acc[3]};
    float4 hi4 = {acc[4], acc[5], acc[6], acc[7]};
    *(float4*)(dst)     = lo4;
    *(float4*)(dst + 4) = hi4;
  }
  __syncthreads();

  // --- per-signal batch OMP (K=4), one thread per signal ---
  float lsum = 0.f;
  if (tid < SIG_PER_WG) {
    int   I[KSPARSE];
    float Lm[10];                   // packed lower-triangular 4x4
    float x[KSPARSE] = {0.f, 0.f, 0.f, 0.f};

    for (int k = 1; k <= KSPARSE; ++k) {
      // argmax_j |h_bar[j] - sum_m x[m]*G[I[m],j]| over unmasked j
      float best = -1.0f; int bidx = 0;
      for (int jj = 0; jj < NUM_ATOMS; ++jj) {
        bool masked = false;
        for (int mm = 0; mm < k - 1; ++mm) masked = masked || (I[mm] == jj);
        if (masked) continue;
        float v = HbT[jj * HBT_STRIDE + tid];
        for (int mm = 0; mm < k - 1; ++mm)
          v -= x[mm] * G[I[mm] * NUM_ATOMS + jj];
        const float av = fabsf(v);
        if (av > best) { best = av; bidx = jj; }
      }
      // incremental Cholesky of G[I,I]
      if (k == 1) {
        Lm[0] = 1.0f;
      } else {
        float w[KSPARSE - 1];
        for (int mm = 0; mm < k - 1; ++mm) {
          float g = G[I[mm] * NUM_ATOMS + bidx];
          for (int cc = 0; cc < mm; ++cc) g -= Lm[mm * (mm + 1) / 2 + cc] * w[cc];
          w[mm] = g / Lm[mm * (mm + 1) / 2 + mm];
        }
        float ss = 0.f;
        for (int mm = 0; mm < k - 1; ++mm) ss += w[mm] * w[mm];
        const float corner = sqrtf(fmaxf(1.0f - ss, 1e-12f));
        const int rr = k - 1;
        for (int cc = 0; cc < k - 1; ++cc) Lm[rr * (rr + 1) / 2 + cc] = w[cc];
        Lm[rr * (rr + 1) / 2 + rr] = corner;
      }
      I[k - 1] = bidx;
      // solve L y = h_bar[I], then L^T x = y
      float y[KSPARSE];
      for (int mm = 0; mm < k; ++mm) {
        float v = HbT[I[mm] * HBT_STRIDE + tid];
        for (int cc = 0; cc < mm; ++cc) v -= Lm[mm * (mm + 1) / 2 + cc] * y[cc];
        y[mm] = v / Lm[mm * (mm + 1) / 2 + mm];
      }
      for (int mm = k - 1; mm >= 0; --mm) {
        float v = y[mm];
        for (int cc = mm + 1; cc < k; ++cc) v -= Lm[cc * (cc + 1) / 2 + mm] * x[cc];
        x[mm] = v / Lm[mm * (mm + 1) / 2 + mm];
      }
    }

    // --- mu-law quantize (N_BINS=16, mu=50, max=3), RTE rounding ---
    const float inv_log1p_mu = 1.0f / log1pf(50.0f);
    float q[KSPARSE];
#pragma unroll
    for (int mm = 0; mm < KSPARSE; ++mm) {
      float c   = fminf(fmaxf(x[mm], -3.0f), 3.0f) * (1.0f / 3.0f);
      float enc = copysignf(log1pf(fabsf(c) * 50.0f) * inv_log1p_mu, c);
      float sc  = (enc + 1.0f) * 7.5f;
      int bin   = (int)rintf(sc);
      bin = bin < 0 ? 0 : (bin > 15 ? 15 : bin);
      float z = (float)bin * (2.0f / 15.0f) - 1.0f;
      q[mm] = copysignf(expm1f(fabsf(z) / inv_log1p_mu) * (1.0f / 50.0f), z) * 3.0f;
    }

    // --- reconstruction, z_q store (lane-coalesced per c), loss partial ---
    for (int c = 0; c < DIM; ++c) {
      float rec = 0.f;
#pragma unroll
      for (int mm = 0; mm < KSPARSE; ++mm)
        rec += q[mm] * Dls[I[mm] * DLS_STRIDE + c];
      out[zbase + (long)c * HW + tid] = rec;       // z_q = recon (forward)
      const float d = rec - Xs[tid * XS_STRIDE + c];
      lsum += d * d;
    }
  }

  // --- block reduction of loss partials: one atomic per workgroup ---
  red[tid] = lsum;
  __syncthreads();
  if (tid == 0) {
    float t = 0.f;
    for (int i = 0; i < 128; ++i) t += red[i];
    // loss = q_latent + 0.25 * e_latent = 1.25 * mean((recon - z_e)^2)
    atomicAdd(out_loss, t * (1.25f / (float)OUT_ZQ_ELEMS));
  }
}

// ---------------------------------------------------------------------------
extern "C" void kernel_launch(void* const* d_in, const int* in_sizes, int n_in,
                              void* d_out, int out_size, void* d_ws, size_t ws_size,
                              hipStream_t stream) {
  (void)in_sizes; (void)n_in; (void)out_size; (void)ws_size;

  const float* ze   = (const float*)d_in[0];   // z_e (32,64,64,64)
  const float* dict = (const float*)d_in[1];   // dictionary (64,256)
  float* out = (float*)d_out;                  // z_q flat (8388608) + loss (1)
  float* ws  = (float*)d_ws;

  float* DnT = ws;                             // 256*64
  float* G   = ws + NUM_ATOMS * DIM;           // 256*256
  float* out_loss = out + OUT_ZQ_ELEMS;

  dl_prep_kernel<<<1, 256, NUM_ATOMS * DIM * sizeof(float), stream>>>(
      dict, DnT, G, out_loss);

  const size_t sh = (size_t)(SIG_PER_WG * XS_STRIDE +
                             NUM_ATOMS * HBT_STRIDE +
                             NUM_ATOMS * DLS_STRIDE + 128) * sizeof(float);
  dl_omp_kernel<<<N_SIGNALS / SIG_PER_WG, 128, sh, stream>>>(
      ze, DnT, G, out, out_loss);
}